// IoULoss_2216203125376
// MI455X (gfx1250) — compile-verified
//
#include <hip/hip_runtime.h>

// ---------------------------------------------------------------------------
// IoU loss (attract + repel) for MI455X / gfx1250, wave32.
// Problem constants from the reference: B=64, N=4096, M=2048, HW=256*256.
// Pipeline:
//   k_attract : 1024 blocks x 256 thr, 1 thread per (b,n) group of 4 corners
//   k_repel   :  512 blocks x 256 thr, 1 thread per (b,m) pair of boxes
//   k_final   : 1 block, deterministic reduction of partials; the last
//               256->1 stage runs on v_wmma_f32_16x16x4_f32 (ones-matmul sum)
// Workspace layout (floats): [0,1024) aLoss | [1024,2048) aCnt
//                            [2048,2560) rLoss | [2560,3072) rCnt
// ---------------------------------------------------------------------------

typedef __attribute__((ext_vector_type(2))) float v2f;
typedef __attribute__((ext_vector_type(8))) float v8f;

#define HW_   65536
#define N_    4096
#define M_    2048

__device__ __forceinline__ float iou3(float hA, float yA, float xA,
                                      float hB, float yB, float xB) {
    float areaA = hA * hA * 0.41f;
    float areaB = hB * hB * 0.41f;
    float ymin = fmaxf(yA - hA * 0.5f,          yB - hB * 0.5f);
    float xmin = fmaxf(xA - 0.41f * hA * 0.5f,  xB - 0.41f * hB * 0.5f);
    float ymax = fminf(yA + hA * 0.5f,          yB + hB * 0.5f);
    float xmax = fminf(xA + 0.41f * hA * 0.5f,  xB + 0.41f * hB * 0.5f);
    float I = fmaxf(ymax - ymin, 0.0f) * fmaxf(xmax - xmin, 0.0f);
    float U = areaA + areaB - I;
    return I / (U + 1e-6f);
}

// Deterministic 256-thread block reduction of two values -> one partial pair.
__device__ __forceinline__ void block_reduce2(float a, float b, int tid,
                                              float* pa, float* pb, int bid) {
    __shared__ float sa[8], sb[8];
    #pragma unroll
    for (int o = 16; o > 0; o >>= 1) {
        a += __shfl_down(a, o, 32);
        b += __shfl_down(b, o, 32);
    }
    int wave = tid >> 5, lane = tid & 31;
    if (lane == 0) { sa[wave] = a; sb[wave] = b; }
    __syncthreads();
    if (tid == 0) {
        float ta = 0.f, tb = 0.f;
        #pragma unroll
        for (int w = 0; w < 8; ++w) { ta += sa[w]; tb += sb[w]; }
        pa[bid] = ta; pb[bid] = tb;
    }
}

__global__ __launch_bounds__(256) void k_attract(
    const float* __restrict__ out_h, const float* __restrict__ out_off,
    const int* __restrict__ attract, const unsigned char* __restrict__ maskA,
    float* __restrict__ pLoss, float* __restrict__ pCnt) {
    const int tid = threadIdx.x;
    const int gid = blockIdx.x * 256 + tid;          // 0 .. B*N-1
    const int b   = gid >> 12;                        // N = 4096

    const int4   i4 = ((const int4*)attract)[gid];    // global_load_b128
    const uchar4 m4 = ((const uchar4*)maskA)[gid];
    const int idx[4] = { i4.x, i4.y, i4.z, i4.w };
    const unsigned char mk[4] = { m4.x, m4.y, m4.z, m4.w };

    const float* hb = out_h  + (size_t)b * HW_;
    const float* ob = out_off + (size_t)b * 2 * HW_;

    float h[4], oy[4], ox[4];
    #pragma unroll
    for (int j = 0; j < 4; ++j) {                     // 12 scattered loads in flight
        h[j]  = hb[idx[j]];
        oy[j] = ob[idx[j]];
        ox[j] = ob[HW_ + idx[j]];
    }

    float hm = 0.f, ym = 0.f, xm = 0.f;
    #pragma unroll
    for (int j = 0; j < 4; ++j) {
        oy[j] += (float)(j & 1);                      // OFFSETS[:,0] = {0,1,0,1}
        ox[j] += (float)((j >> 1) & 1);               // OFFSETS[:,1] = {0,0,1,1}
        hm += h[j]; ym += oy[j]; xm += ox[j];
    }
    hm *= 0.25f; ym *= 0.25f; xm *= 0.25f;
    const float hB = expf(hm);

    float ll = 0.f, lc = 0.f;
    #pragma unroll
    for (int j = 0; j < 4; ++j) {
        float v = iou3(expf(h[j]), oy[j], ox[j], hB, ym, xm);
        if (mk[j]) { ll += 1.0f - v; lc += 1.0f; }
    }
    block_reduce2(ll, lc, tid, pLoss, pCnt, blockIdx.x);
}

__global__ __launch_bounds__(256) void k_repel(
    const float* __restrict__ out_h, const float* __restrict__ out_off,
    const int* __restrict__ repel, const float* __restrict__ pre_off,
    const unsigned char* __restrict__ maskR,
    float* __restrict__ pLoss, float* __restrict__ pCnt) {
    const int tid = threadIdx.x;
    const int gid = blockIdx.x * 256 + tid;          // 0 .. B*M-1
    const int b   = gid >> 11;                        // M = 2048

    const float* hb = out_h  + (size_t)b * HW_;
    const float* ob = out_off + (size_t)b * 2 * HW_;

    float hmean[2], oymean[2], oxmean[2];
    #pragma unroll
    for (int p = 0; p < 2; ++p) {
        const int4 i4 = ((const int4*)repel)[gid * 2 + p];
        const int idx[4] = { i4.x, i4.y, i4.z, i4.w };
        float hs = 0.f, ys = 0.f, xs = 0.f;
        #pragma unroll
        for (int j = 0; j < 4; ++j) {
            hs += hb[idx[j]];
            ys += ob[idx[j]]       + (float)(j & 1);
            xs += ob[HW_ + idx[j]] + (float)((j >> 1) & 1);
        }
        hmean[p] = hs * 0.25f; oymean[p] = ys * 0.25f; oxmean[p] = xs * 0.25f;
    }
    const float2 po = ((const float2*)pre_off)[gid];  // pre_off[b,m,0:2]
    oymean[1] += po.x;
    oxmean[1] += po.y;

    float v = iou3(expf(hmean[0]), oymean[0], oxmean[0],
                   expf(hmean[1]), oymean[1], oxmean[1]);

    float rl = 0.f, rc = 0.f;
    if (maskR[gid]) { rl = v; rc = 1.0f; }
    block_reduce2(rl, rc, tid, pLoss, pCnt, blockIdx.x);
}

__global__ __launch_bounds__(256) void k_final(const float* __restrict__ ws,
                                               float* __restrict__ out) {
    __shared__ float s[256];
    __shared__ float res[4];
    const int tid = threadIdx.x;
    const float* aL = ws;
    const float* aC = ws + 1024;
    const float* rL = ws + 2048;
    const float* rC = ws + 2560;

    float vals[4];
    vals[0] = aL[tid] + aL[tid + 256] + aL[tid + 512] + aL[tid + 768];
    vals[1] = aC[tid] + aC[tid + 256] + aC[tid + 512] + aC[tid + 768];
    vals[2] = rL[tid] + rL[tid + 256];
    vals[3] = rC[tid] + rC[tid + 256];

    #pragma unroll
    for (int q = 0; q < 4; ++q) {
        __syncthreads();
        s[tid] = vals[q];
        __syncthreads();
        if (tid < 32) {                               // wave 0, EXEC all ones
            // Sum 256 f32 via ones-matmul: D += ones(16x4) x B(4x16).
            // Any bijective lane->(k,n) packing of each 64-value chunk is
            // valid since every weight is 1.0; 4 chained WMMAs cover 256.
            v8f c = {};
            v2f a; a.x = 1.0f; a.y = 1.0f;
            #pragma unroll
            for (int chunk = 0; chunk < 4; ++chunk) {
                v2f bb;
                bb.x = s[chunk * 64 + tid];
                bb.y = s[chunk * 64 + 32 + tid];
                c = __builtin_amdgcn_wmma_f32_16x16x4_f32(
                        /*neg_a=*/false, a, /*neg_b=*/false, bb,
                        /*c_mod=*/(short)0, c,
                        /*reuse_a=*/false, /*reuse_b=*/false);
            }
            // c[0], lanes 0..15 = row M=0 -> the 16 column sums.
            float v = c[0];
            v += __shfl_xor(v, 1, 32);
            v += __shfl_xor(v, 2, 32);
            v += __shfl_xor(v, 4, 32);
            v += __shfl_xor(v, 8, 32);
            if (tid == 0) res[q] = v;
        }
    }
    __syncthreads();
    if (tid == 0) {
        out[0] = res[0] / (res[1] + 1e-4f) + res[2] / (res[3] + 1e-4f);
    }
}

extern "C" void kernel_launch(void* const* d_in, const int* in_sizes, int n_in,
                              void* d_out, int out_size, void* d_ws, size_t ws_size,
                              hipStream_t stream) {
    (void)in_sizes; (void)n_in; (void)out_size; (void)ws_size;
    const float*         out_h   = (const float*)d_in[0];
    const float*         out_off = (const float*)d_in[1];
    // d_in[2]=target_h, d_in[3]=target_off: unused by the reference.
    const float*         pre_off = (const float*)d_in[4];
    const int*           attract = (const int*)d_in[5];
    const int*           repel   = (const int*)d_in[6];
    const unsigned char* maskA   = (const unsigned char*)d_in[7];
    const unsigned char* maskR   = (const unsigned char*)d_in[8];

    float* ws = (float*)d_ws;   // 3072 floats = 12 KB, all slots overwritten

    k_attract<<<1024, 256, 0, stream>>>(out_h, out_off, attract, maskA,
                                        ws, ws + 1024);
    k_repel<<<512, 256, 0, stream>>>(out_h, out_off, repel, pre_off, maskR,
                                     ws + 2048, ws + 2560);
    k_final<<<1, 256, 0, stream>>>(ws, (float*)d_out);
}